// KMeans_Get_Cluster_Index_and_Centroids_4638564679950
// MI455X (gfx1250) — compile-verified
//
#include <hip/hip_runtime.h>
#include <stdint.h>
#include <stddef.h>

#define BATCH 4
#define NPTS  8192
#define DIM   256
#define KCL   512
#define NITER 10

#define MTILE 128          // points per workgroup
#define LDS_STRIDE 260     // floats per row in LDS (256 + 4 pad -> conflict-free A reads)

typedef __attribute__((ext_vector_type(2))) float v2f;
typedef __attribute__((ext_vector_type(4))) float v4f;
typedef __attribute__((ext_vector_type(8))) float v8f;
typedef __attribute__((ext_vector_type(4))) unsigned int u32x4;
typedef __attribute__((ext_vector_type(4))) int i32x4;
typedef __attribute__((ext_vector_type(8))) int i32x8;

#if defined(__has_builtin)
#if __has_builtin(__builtin_amdgcn_tensor_load_to_lds) && __has_builtin(__builtin_amdgcn_s_wait_tensorcnt)
#define HAVE_TDM 1
#endif
#endif
#ifndef HAVE_TDM
#define HAVE_TDM 0
#endif

#if HAVE_TDM
#if __has_include(<hip/amd_detail/amd_gfx1250_TDM.h>)
#define TDM_6ARG 1
#pragma message("CDNA5 probe: TDM path active, 6-arg tensor_load_to_lds (amdgpu-toolchain headers)")
#else
#define TDM_6ARG 0
#pragma message("CDNA5 probe: TDM path active, 5-arg tensor_load_to_lds (ROCm toolchain)")
#endif
#else
#pragma message("CDNA5 probe: TDM builtin NOT available, manual LDS staging fallback")
#endif

// ---------------------------------------------------------------------------
// Assign: for each point, argmin_k (c2[k] - 2 * dot(x_n, c_k)).
// One workgroup = 8 waves = 128 points. Each wave computes 16 pts x 32
// clusters per pass: two independent V_WMMA_F32_16X16X4_F32 accumulation
// chains share every A fragment (halves LDS reads, hides WMMA latency).
// x-tile staged into LDS via the Tensor Data Mover (padded rows, stride 260).
// ---------------------------------------------------------------------------
__global__ __launch_bounds__(256) void kmeans_assign(
    const float* __restrict__ x, const float* __restrict__ centers,
    const float* __restrict__ c2, float* __restrict__ labels)
{
  __shared__ float xs[MTILE * LDS_STRIDE];   // 130 KB of the 320 KB WGP LDS

  const int b    = blockIdx.y;
  const int m0   = blockIdx.x * MTILE;
  const int tid  = threadIdx.x;
  const int lane = tid & 31;
  const int wave = tid >> 5;
  const int hl   = lane >> 4;      // half-wave: selects K sub-pair of the frag
  const int l15  = lane & 15;

  const float* xb = x + (size_t)b * NPTS * DIM;

#if HAVE_TDM
  if (tid < 32) {
    uint64_t ga   = (uint64_t)(uintptr_t)(const void*)(xb + (size_t)m0 * DIM);
    unsigned ldsb = (unsigned)(uintptr_t)(void*)xs;
    u32x4 g0;
    g0.x = 1u;                                     // count=1, user descriptor
    g0.y = ldsb;                                   // lds_addr
    g0.z = (unsigned)ga;                           // global_addr[31:0]
    g0.w = (unsigned)((ga >> 32) & 0x01FFFFFFu) | (2u << 30); // addr[56:32] | type=2
    i32x8 g1;
    g1.s0 = (2 << 16) | (1 << 20) | (7 << 22) | (3 << 25); // 4B elems, pad_en, every 256dw pad 4dw
    g1.s1 = (DIM & 0xFFFF) << 16;                  // tensor_dim0 = 256 (lo16)
    g1.s2 = (NPTS & 0xFFFF) << 16;                 // tensor_dim1 = 8192 (lo16)
    g1.s3 = (DIM & 0xFFFF) << 16;                  // tile_dim0 = 256
    g1.s4 = MTILE;                                 // tile_dim1 = 128, tile_dim2 = 0
    g1.s5 = DIM;                                   // tensor_dim0_stride = 256
    g1.s6 = 0;
    g1.s7 = 0;
    i32x4 z4 = (i32x4)0;
#if TDM_6ARG
    i32x8 z8 = (i32x8)0;
    __builtin_amdgcn_tensor_load_to_lds(g0, g1, z4, z4, z8, 0);
#else
    __builtin_amdgcn_tensor_load_to_lds(g0, g1, z4, z4, 0);
#endif
    __builtin_amdgcn_s_wait_tensorcnt(0);
  }
  __syncthreads();
#else
  for (int i = tid; i < MTILE * (DIM / 4); i += 256) {
    int row = i >> 6, c4 = i & 63;
    v4f v = *(const v4f*)(xb + (size_t)(m0 + row) * DIM + c4 * 4);
    *(v4f*)(&xs[row * LDS_STRIDE + c4 * 4]) = v;
  }
  __syncthreads();
#endif

  const int    rowA = wave * 16 + l15;             // A-frag row (point)
  const float* cb   = centers + (size_t)b * KCL * DIM;
  const float* c2b  = c2 + b * KCL;
  const float* Aptr = &xs[rowA * LDS_STRIDE + 2 * hl];

  float best[8];
  int   bidx[8];
  #pragma unroll
  for (int j = 0; j < 8; ++j) { best[j] = 3.4e38f; bidx[j] = 0; }

  for (int kt = 0; kt < KCL / 32; ++kt) {
    const int    ncl0 = kt * 32 + l15;             // B-frag columns (clusters)
    const int    ncl1 = ncl0 + 16;
    const float  c2v0 = c2b[ncl0];
    const float  c2v1 = c2b[ncl1];
    const float* B0   = cb + (size_t)ncl0 * DIM + 2 * hl;
    const float* B1   = cb + (size_t)ncl1 * DIM + 2 * hl;

    v8f acc0 = {0.f, 0.f, 0.f, 0.f, 0.f, 0.f, 0.f, 0.f};
    v8f acc1 = {0.f, 0.f, 0.f, 0.f, 0.f, 0.f, 0.f, 0.f};
    #pragma unroll 4
    for (int kk = 0; kk < DIM; kk += 4) {
      v2f a  = *(const v2f*)(Aptr + kk);           // A 16x4 f32 fragment (shared)
      v2f b0 = *(const v2f*)(B0 + kk);             // B 4x16 f32 fragments
      v2f b1 = *(const v2f*)(B1 + kk);
      acc0 = __builtin_amdgcn_wmma_f32_16x16x4_f32(
          false, a, false, b0, (short)0, acc0, false, false);
      acc1 = __builtin_amdgcn_wmma_f32_16x16x4_f32(
          false, a, false, b1, (short)0, acc1, false, false);
    }
    #pragma unroll
    for (int j = 0; j < 8; ++j) {
      float s0 = c2v0 - 2.0f * acc0[j];
      if (s0 < best[j]) { best[j] = s0; bidx[j] = ncl0; }
      float s1 = c2v1 - 2.0f * acc1[j];
      if (s1 < best[j]) { best[j] = s1; bidx[j] = ncl1; }
    }
  }

  // reduce argmin across the 16 lanes holding different clusters (wave32)
  #pragma unroll
  for (int m = 1; m < 16; m <<= 1) {
    #pragma unroll
    for (int j = 0; j < 8; ++j) {
      float ov = __shfl_xor(best[j], m, 32);
      int   oi = __shfl_xor(bidx[j], m, 32);
      if (ov < best[j] || (ov == best[j] && oi < bidx[j])) { best[j] = ov; bidx[j] = oi; }
    }
  }
  // lane 0 holds rows 0..7 of this wave's 16-row slab, lane 16 holds rows 8..15
  if (l15 == 0) {
    float* lb = labels + (size_t)b * NPTS + m0 + wave * 16 + hl * 8;
    #pragma unroll
    for (int j = 0; j < 8; ++j) lb[j] = (float)bidx[j];
  }
}

// c2[k] = sum_d centers[k][d]^2 ; one wave32 per cluster row
__global__ __launch_bounds__(256) void kmeans_c2(const float* __restrict__ centers,
                                                 float* __restrict__ c2)
{
  int lane = threadIdx.x & 31, wave = threadIdx.x >> 5;
  int gk = blockIdx.x * 8 + wave;                  // 0 .. BATCH*KCL-1
  const float* row = centers + (size_t)gk * DIM;
  float s = 0.f;
  #pragma unroll
  for (int i = 0; i < 8; ++i) { float v = row[lane + 32 * i]; s += v * v; }
  #pragma unroll
  for (int m = 1; m < 32; m <<= 1) s += __shfl_xor(s, m, 32);
  if (lane == 0) c2[gk] = s;
}

__global__ __launch_bounds__(256) void kmeans_zero(float* __restrict__ sums,
                                                   float* __restrict__ counts)
{
  int i = blockIdx.x * blockDim.x + threadIdx.x;   // covers BATCH*KCL*DIM
  sums[i] = 0.f;
  if (i < BATCH * KCL) counts[i] = 0.f;
}

// one wave32 per point: 8 coalesced 32-lane f32 atomics into its cluster row
__global__ __launch_bounds__(256) void kmeans_accum(
    const float* __restrict__ x, const float* __restrict__ labels,
    float* __restrict__ sums, float* __restrict__ counts)
{
  int lane = threadIdx.x & 31, wave = threadIdx.x >> 5;
  int b = blockIdx.y;
  int n = blockIdx.x * 8 + wave;
  int lbl = (int)labels[(size_t)b * NPTS + n];
  const float* xr = x + ((size_t)b * NPTS + n) * DIM;
  float* sr = sums + ((size_t)b * KCL + lbl) * DIM;
  #pragma unroll
  for (int i = 0; i < 8; ++i)
    atomicAdd(&sr[lane + 32 * i], xr[lane + 32 * i]);
  if (lane == 0) atomicAdd(&counts[b * KCL + lbl], 1.0f);
}

__global__ __launch_bounds__(256) void kmeans_update(
    const float* __restrict__ sums, const float* __restrict__ counts,
    float* __restrict__ centers)
{
  int i = blockIdx.x * blockDim.x + threadIdx.x;   // covers BATCH*KCL*DIM
  int bk = i >> 8;
  float c = counts[bk];
  if (c > 0.f) centers[i] = sums[i] / c;           // else keep old center
}

__global__ __launch_bounds__(256) void kmeans_init(const float* __restrict__ x,
                                                   float* __restrict__ centers)
{
  int i = blockIdx.x * blockDim.x + threadIdx.x;   // covers BATCH*KCL*DIM
  int b = i / (KCL * DIM);
  int rem = i - b * (KCL * DIM);
  centers[i] = x[(size_t)b * NPTS * DIM + rem];    // centers0 = x[:, :512, :]
}

extern "C" void kernel_launch(void* const* d_in, const int* in_sizes, int n_in,
                              void* d_out, int out_size, void* d_ws, size_t ws_size,
                              hipStream_t stream)
{
  (void)in_sizes; (void)n_in; (void)out_size; (void)ws_size;
  const float* x = (const float*)d_in[0];

  float* out     = (float*)d_out;
  float* labels  = out;                        // 4*8192 (written as float values)
  float* centers = out + BATCH * NPTS;         // 4*512*256 == centers output

  float* c2     = (float*)d_ws;                           // 4*512
  float* sums   = c2 + BATCH * KCL;                       // 4*512*256
  float* counts = sums + (size_t)BATCH * KCL * DIM;       // 4*512

  dim3 blk(256);
  kmeans_init<<<BATCH * KCL * DIM / 256, blk, 0, stream>>>(x, centers);
  for (int it = 0; it < NITER; ++it) {
    kmeans_c2<<<BATCH * KCL / 8, blk, 0, stream>>>(centers, c2);
    kmeans_zero<<<BATCH * KCL * DIM / 256, blk, 0, stream>>>(sums, counts);
    kmeans_assign<<<dim3(NPTS / MTILE, BATCH), blk, 0, stream>>>(x, centers, c2, labels);
    kmeans_accum<<<dim3(NPTS / 8, BATCH), blk, 0, stream>>>(x, labels, sums, counts);
    kmeans_update<<<BATCH * KCL * DIM / 256, blk, 0, stream>>>(sums, counts, centers);
  }
  kmeans_c2<<<BATCH * KCL / 8, blk, 0, stream>>>(centers, c2);
  kmeans_assign<<<dim3(NPTS / MTILE, BATCH), blk, 0, stream>>>(x, centers, c2, labels);
}